// CategorySpecificSequenceMLP_82592221102550
// MI455X (gfx1250) — compile-verified
//
#include <hip/hip_runtime.h>
#include <hip/hip_bf16.h>

typedef __attribute__((ext_vector_type(2))) float v2f;
typedef __attribute__((ext_vector_type(8))) float v8f;

#define NUM_CAT   8
#define BATCH     32
#define IN_DIM    4096
#define MID_DIM_  1024
#define OUT_DIM_  512
#define KSPLIT    4

#define DEPTH        8               // panels in flight per ring (A ring + B ring)
#define PANEL_FLOATS 128             // 512 B: A: 16 rows x 8 K ; B: 8 K-rows x 16 cols
#define PANEL_BYTES  512

// ---- CDNA5 async-to-LDS helpers (inline asm; no builtin on this toolchain) ----
__device__ __forceinline__ void async_b128_nt(unsigned lds_off, const float* g) {
    asm volatile("global_load_async_to_lds_b128 %0, %1, off th:TH_LOAD_NT"
                 :: "v"(lds_off), "v"(g) : "memory");
}
__device__ __forceinline__ void async_b128_rt(unsigned lds_off, const float* g) {
    asm volatile("global_load_async_to_lds_b128 %0, %1, off"
                 :: "v"(lds_off), "v"(g) : "memory");
}
template <int N> __device__ __forceinline__ void wait_asynccnt() {
    asm volatile("s_wait_asynccnt %0" :: "n"(N) : "memory");
}
__device__ __forceinline__ void wait_dscnt0() {
    asm volatile("s_wait_dscnt 0" ::: "memory");
}

// ---------------------------------------------------------------------------
// Init: zero scratch row + per-category sample lists. Deterministic.
// ---------------------------------------------------------------------------
__global__ void init_kernel(const int* __restrict__ cat_ids,
                            int* __restrict__ counts,
                            int* __restrict__ idx,
                            float* __restrict__ zrow) {
    for (int i = threadIdx.x; i < IN_DIM; i += blockDim.x) zrow[i] = 0.0f;
    if (threadIdx.x == 0) {
        for (int c = 0; c < NUM_CAT; ++c) counts[c] = 0;
        for (int b = 0; b < BATCH; ++b) {
            int c = cat_ids[b];
            idx[c * BATCH + counts[c]] = b;
            counts[c]++;
        }
    }
}

// ---------------------------------------------------------------------------
// Category GEMM partial (K-split) with V_WMMA_F32_16X16X4_F32.
// BOTH operands stream via GLOBAL_LOAD_ASYNC_TO_LDS_B128 into wave-private
// LDS rings (ASYNCcnt-tracked). Inner loop has no synchronous global loads.
// grid: (N/64, 8 cats, 2 m-tiles * KSPLIT), block 128 (4 waves).
// ---------------------------------------------------------------------------
template <int K_DIM, int N_DIM, int S>
__global__ __launch_bounds__(128) void cat_gemm_part(
    const float* __restrict__ A,     // [BATCH, K_DIM]
    const float* __restrict__ W,     // [NUM_CAT, K_DIM, N_DIM]
    const int*   __restrict__ counts,
    const int*   __restrict__ idx,   // [NUM_CAT, BATCH]
    const float* __restrict__ zrow,  // >= K_DIM zeros
    float*       __restrict__ Part)  // [S, BATCH, N_DIM]
{
    const int c      = blockIdx.y;
    const int mtile  = blockIdx.z / S;
    const int ks     = blockIdx.z % S;
    const int count  = counts[c];
    const int mstart = mtile * 16;
    if (mstart >= count) return;                 // uniform early-out (before barrier)

    const int wave  = threadIdx.x >> 5;
    const int lane  = threadIdx.x & 31;
    const int ncol0 = blockIdx.x * 64 + wave * 16;

    __shared__ int   sidx[16];
    __shared__ float smem[4 * 2 * DEPTH * PANEL_FLOATS];   // 32 KB: A+B rings x 4 waves

    if (threadIdx.x < 16) {
        int m = mstart + threadIdx.x;
        sidx[threadIdx.x] = (m < count) ? idx[c * BATCH + m] : -1;
    }
    __syncthreads();

    const int col = lane & 15;                   // B/D column owned by this lane
    const int kb  = (lane >> 4) * 2;             // A/B K sub-offset per lane half
    const int k0  = ks * (K_DIM / S);
    const int NPANEL = (K_DIM / S) / 8;

    // ---- async source addresses (per lane, 16B each; panel = 512B/wave-op) ----
    // A panel: lane L -> row (L>>1) [scattered by sample], K chunk (L&1)*4.
    const int    arow_i = lane >> 1;
    const int    asamp  = sidx[arow_i];
    const float* gA = (asamp >= 0 ? A + (size_t)asamp * K_DIM : zrow)
                      + k0 + (lane & 1) * 4;
    // B panel: lane L -> W row k0+(L>>2), cols ncol0+(L&3)*4.
    const float* gB = W + (size_t)c * K_DIM * N_DIM
                      + (size_t)(k0 + (lane >> 2)) * N_DIM + ncol0 + (lane & 3) * 4;

    float* aring = &smem[(wave * 2 + 0) * DEPTH * PANEL_FLOATS];
    float* bring = &smem[(wave * 2 + 1) * DEPTH * PANEL_FLOATS];
    const unsigned ldsA = (unsigned)(size_t)aring + (unsigned)(lane * 16);
    const unsigned ldsB = (unsigned)(size_t)bring + (unsigned)(lane * 16);

    // ---- prologue: fill both rings (issue order A_j then B_j) ----
#pragma unroll
    for (int j = 0; j < DEPTH; ++j) {
        async_b128_rt(ldsA + j * PANEL_BYTES, gA);  gA += 8;
        async_b128_nt(ldsB + j * PANEL_BYTES, gB);  gB += 8 * (size_t)N_DIM;
    }

    v8f acc = {};
    // ---- main loop: branchless, unconditional re-issue ----
    for (int j = 0; j < NPANEL - DEPTH; ++j) {
        wait_asynccnt<2 * DEPTH - 2>();          // in-order: A_j and B_j resident
        const int slot = j & (DEPTH - 1);
        const float* pa = &aring[slot * PANEL_FLOATS + (lane & 15) * 8];
        const float* pb = &bring[slot * PANEL_FLOATS];
        v2f av0 = *(const v2f*)(pa + kb);
        v2f av1 = *(const v2f*)(pa + 4 + kb);
        v2f bv0, bv1;
        bv0.x = pb[kb * 16 + col];       bv0.y = pb[(kb + 1) * 16 + col];
        bv1.x = pb[(4 + kb) * 16 + col]; bv1.y = pb[(5 + kb) * 16 + col];
        wait_dscnt0();                           // slot reads retired before rewrite
        async_b128_rt(ldsA + slot * PANEL_BYTES, gA);  gA += 8;
        async_b128_nt(ldsB + slot * PANEL_BYTES, gB);  gB += 8 * (size_t)N_DIM;
        acc = __builtin_amdgcn_wmma_f32_16x16x4_f32(false, av0, false, bv0,
                                                    (short)0, acc, false, false);
        acc = __builtin_amdgcn_wmma_f32_16x16x4_f32(false, av1, false, bv1,
                                                    (short)0, acc, false, false);
    }

    // ---- epilogue: drain once, then consume remaining DEPTH panels wait-free ----
    wait_asynccnt<0>();
#pragma unroll
    for (int e = 0; e < DEPTH; ++e) {
        const int j = NPANEL - DEPTH + e;
        const int slot = j & (DEPTH - 1);
        const float* pa = &aring[slot * PANEL_FLOATS + (lane & 15) * 8];
        const float* pb = &bring[slot * PANEL_FLOATS];
        v2f av0 = *(const v2f*)(pa + kb);
        v2f av1 = *(const v2f*)(pa + 4 + kb);
        v2f bv0, bv1;
        bv0.x = pb[kb * 16 + col];       bv0.y = pb[(kb + 1) * 16 + col];
        bv1.x = pb[(4 + kb) * 16 + col]; bv1.y = pb[(5 + kb) * 16 + col];
        acc = __builtin_amdgcn_wmma_f32_16x16x4_f32(false, av0, false, bv0,
                                                    (short)0, acc, false, false);
        acc = __builtin_amdgcn_wmma_f32_16x16x4_f32(false, av1, false, bv1,
                                                    (short)0, acc, false, false);
    }

    // D layout: VGPR r -> M = r (+8 for lanes 16-31), column = lane&15.
    const int mhi = (lane >> 4) * 8;
#pragma unroll
    for (int r = 0; r < 8; ++r) {
        int s = sidx[r + mhi];
        if (s >= 0)
            Part[(size_t)ks * (BATCH * N_DIM) + (size_t)s * N_DIM + ncol0 + col] = acc[r];
    }
}

// ---------------------------------------------------------------------------
// Reduce K-split partials + bias (+ optional ReLU). Fixed order -> deterministic.
// ---------------------------------------------------------------------------
template <int S, int N, bool RELU>
__global__ __launch_bounds__(256) void reduce_bias_kernel(
    const float* __restrict__ part,     // [S, BATCH, N]
    const float* __restrict__ bias,     // [NUM_CAT, N]
    const int*   __restrict__ cat_ids,  // [BATCH]
    float*       __restrict__ out)      // [BATCH, N]
{
    int i = blockIdx.x * blockDim.x + threadIdx.x;
    if (i >= BATCH * N) return;
    int sample = i / N;
    int col    = i - sample * N;
    float s = 0.0f;
#pragma unroll
    for (int j = 0; j < S; ++j) s += part[(size_t)j * (BATCH * N) + i];
    s += bias[cat_ids[sample] * N + col];
    if (RELU) s = s > 0.0f ? s : 0.0f;
    out[i] = s;
}

// ---------------------------------------------------------------------------
// Inputs (setup_inputs order): x, W1, b1, W2, b2, cat_ids. Output: [32,16,32] f32.
// ---------------------------------------------------------------------------
extern "C" void kernel_launch(void* const* d_in, const int* in_sizes, int n_in,
                              void* d_out, int out_size, void* d_ws, size_t ws_size,
                              hipStream_t stream) {
    const float* x       = (const float*)d_in[0];
    const float* W1      = (const float*)d_in[1];
    const float* b1      = (const float*)d_in[2];
    const float* W2      = (const float*)d_in[3];
    const float* b2      = (const float*)d_in[4];
    const int*   cat_ids = (const int*)  d_in[5];
    float*       out     = (float*)d_out;

    // workspace layout (bytes):
    //      0: counts[8]        |     64: idx[8*32]
    //   4096: zrow[4096 f32]   |  32768: H[32*1024 f32]
    // 163840: Hpart[4][32][1024] f32 (512 KB)
    // 688128: Opart[4][32][512]  f32 (256 KB)
    char*  ws     = (char*)d_ws;
    int*   counts = (int*)ws;
    int*   idx    = (int*)(ws + 64);
    float* zrow   = (float*)(ws + 4096);
    float* H      = (float*)(ws + 32768);
    float* Hpart  = (float*)(ws + 163840);
    float* Opart  = (float*)(ws + 688128);

    init_kernel<<<1, 256, 0, stream>>>(cat_ids, counts, idx, zrow);

    // layer1 partials: [B,4096] x [4096,1024], K split 4 ways
    cat_gemm_part<IN_DIM, MID_DIM_, KSPLIT>
        <<<dim3(MID_DIM_ / 64, NUM_CAT, 2 * KSPLIT), 128, 0, stream>>>(
            x, W1, counts, idx, zrow, Hpart);
    reduce_bias_kernel<KSPLIT, MID_DIM_, true>
        <<<(BATCH * MID_DIM_ + 255) / 256, 256, 0, stream>>>(
            Hpart, b1, cat_ids, H);

    // layer2 partials: [B,1024] x [1024,512], K split 4 ways
    cat_gemm_part<MID_DIM_, OUT_DIM_, KSPLIT>
        <<<dim3(OUT_DIM_ / 64, NUM_CAT, 2 * KSPLIT), 128, 0, stream>>>(
            H, W2, counts, idx, zrow, Opart);
    reduce_bias_kernel<KSPLIT, OUT_DIM_, false>
        <<<(BATCH * OUT_DIM_ + 255) / 256, 256, 0, stream>>>(
            Opart, b2, cat_ids, out);
}